// NeuralNetwork_27745488732940
// MI455X (gfx1250) — compile-verified
//
#include <hip/hip_runtime.h>
#include <cstdint>
#include <cstddef>

// ---------------- problem constants (match reference) ----------------
#define KK 512                  // edges per target
#define KSPLIT 4                // waves cooperating on one 16-target group
#define KQ (KK / KSPLIT)        // 128 edges per wave per target
#define TGT_PER_GROUP 16
#define GROUPS_PER_BLOCK 2
#define WAVES_PER_BLOCK (KSPLIT * GROUPS_PER_BLOCK)      // 8
#define BLOCK_THREADS (WAVES_PER_BLOCK * 32)             // 256
#define TGT_PER_BLOCK (TGT_PER_GROUP * GROUPS_PER_BLOCK) // 32
#define MAX_SRC 4096

typedef float v2f __attribute__((ext_vector_type(2)));
typedef float v4f __attribute__((ext_vector_type(4)));
typedef float v8f __attribute__((ext_vector_type(8)));
typedef unsigned int v4u __attribute__((ext_vector_type(4)));
typedef int v8i __attribute__((ext_vector_type(8)));
typedef int v4i __attribute__((ext_vector_type(4)));

#if __has_builtin(__builtin_amdgcn_tensor_load_to_lds) && __has_builtin(__builtin_amdgcn_s_wait_tensorcnt)
#define HAS_TDM 1
#else
#define HAS_TDM 0
#endif
#if __has_builtin(__builtin_amdgcn_wmma_f32_16x16x4_f32)
#define HAS_WMMA_F32X4 1
#else
#define HAS_WMMA_F32X4 0
#endif

// One kernel per level: dst[t] = sum_{e in t's K edges} w[e] * src_state[ef[e] - src_lo]
// Last level additionally applies tanh(. + bias).
__global__ __launch_bounds__(BLOCK_THREADS)
void nn_level_kernel(const float* __restrict__ src, const float* __restrict__ wts,
                     const int* __restrict__ ef, float* __restrict__ dst,
                     const float* __restrict__ bias, int src_lo, int src_n)
{
    __shared__ float s_src[MAX_SRC];                                  // 16 KB source level
    __shared__ float s_red[GROUPS_PER_BLOCK][TGT_PER_GROUP][KSPLIT * 32]; // 16 KB partials

    const int tid   = threadIdx.x;
    const int lane  = tid & 31;
    const int wave  = tid >> 5;
    const int q     = wave & (KSPLIT - 1);   // which K-quarter this wave owns
    const int group = wave / KSPLIT;         // which 16-target group

#if HAS_TDM
    // ---- Tensor Data Mover: DMA src_n floats (<=16KB) global -> LDS ----
    if (tid == 0) {
        uint64_t ga = (uint64_t)(uintptr_t)src;
        uint32_t lds_addr = (uint32_t)(uintptr_t)(&s_src[0]); // low 32 bits = LDS offset
        v4u g0 = { 1u,                                        // count=1, user mode
                   lds_addr,                                  // lds_addr [63:32]
                   (uint32_t)ga,                              // global_addr lo
                   (uint32_t)((ga >> 32) & 0x1FFFFFFull) | (2u << 30) }; // addr hi + type=2
        uint32_t n = (uint32_t)src_n;
        v8i g1 = { (int)0x00020000u,                          // data_size=4B, mask=0
                   (int)((n & 0xFFFFu) << 16),                // tensor_dim0 lo16 @ [63:48]
                   (int)((1u << 16) | (n >> 16)),             // tensor_dim1=1, tdim0 hi16
                   (int)((n & 0xFFFFu) << 16),                // tile_dim0 @ [127:112]
                   1,                                         // tile_dim1=1
                   (int)n,                                    // tensor_dim0_stride lo32
                   0, 0 };
        v4i g2 = {0,0,0,0};
        v4i g3 = {0,0,0,0};
        v8i g4 = {0,0,0,0,0,0,0,0};
        __builtin_amdgcn_tensor_load_to_lds(g0, g1, g2, g3, g4, 0);
        __builtin_amdgcn_s_wait_tensorcnt(0);
    }
#else
    for (int i = tid; i < (src_n >> 2); i += BLOCK_THREADS)
        ((float4*)s_src)[i] = ((const float4*)src)[i];
#endif
    __syncthreads();

    const int t_base = blockIdx.x * TGT_PER_BLOCK + group * TGT_PER_GROUP;

    // ---- stream weights + indices (coalesced b128, non-temporal), LDS gather, FMA ----
    // wave handles edges [q*128, q*128+128) of each of its 16 targets: 1 float4 + 1 int4 per lane
    float acc[TGT_PER_GROUP];
#pragma unroll 4
    for (int m = 0; m < TGT_PER_GROUP; ++m) {
        const int t = t_base + m;
        const v4f* wrow = (const v4f*)(wts + (size_t)t * KK) + q * 32 + lane;
        const v4i* irow = (const v4i*)(ef  + (size_t)t * KK) + q * 32 + lane;
        if (m + 1 < TGT_PER_GROUP) {
            __builtin_prefetch(wts + (size_t)(t + 1) * KK + q * KQ + lane * 4, 0, 0);
            __builtin_prefetch(ef  + (size_t)(t + 1) * KK + q * KQ + lane * 4, 0, 0);
        }
        v4f w = __builtin_nontemporal_load(wrow);
        v4i ix = __builtin_nontemporal_load(irow);
        float a;
        a  = w.x * s_src[ix.x - src_lo];
        a += w.y * s_src[ix.y - src_lo];
        a += w.z * s_src[ix.z - src_lo];
        a += w.w * s_src[ix.w - src_lo];
        acc[m] = a;
    }

    // deposit partials: s_red[group][m][q*32+lane]
#pragma unroll
    for (int m = 0; m < TGT_PER_GROUP; ++m) s_red[group][m][q * 32 + lane] = acc[m];
    __syncthreads();

    // one wave per group reduces its 16x128 partial tile
    if (q == 0) {
#if HAS_WMMA_F32X4
        // ---- WMMA 16x16x4 f32 as a 16x128 -> 16 cross-lane reducer (B = ones) ----
        const int row  = lane & 15;
        const int half = lane >> 4;
        v2f bmat; bmat.x = 1.0f; bmat.y = 1.0f;
        v8f cacc = {0.f,0.f,0.f,0.f,0.f,0.f,0.f,0.f};
#pragma unroll
        for (int kb = 0; kb < KSPLIT * 32; kb += 4) {
            v2f amat;
            amat.x = s_red[group][row][kb + half * 2 + 0];
            amat.y = s_red[group][row][kb + half * 2 + 1];
            cacc = __builtin_amdgcn_wmma_f32_16x16x4_f32(false, amat, false, bmat,
                                                         (short)0, cacc, false, false);
        }
        // D layout: VGPR j of lanes 0-15 -> M=j ; lanes 16-31 -> M=8+j (all N cols equal)
        if (row == 0) {
#pragma unroll
            for (int j = 0; j < 8; ++j) {
                int t = t_base + half * 8 + j;
                float v = cacc[j];
                dst[t] = bias ? tanhf(v + bias[t]) : v;
            }
        }
#else
        if (lane < TGT_PER_GROUP) {
            float v = 0.0f;
#pragma unroll
            for (int l = 0; l < KSPLIT * 32; ++l) v += s_red[group][lane][l];
            int t = t_base + lane;
            dst[t] = bias ? tanhf(v + bias[t]) : v;
        }
#endif
    }
}

extern "C" void kernel_launch(void* const* d_in, const int* in_sizes, int n_in,
                              void* d_out, int out_size, void* d_ws, size_t ws_size,
                              hipStream_t stream) {
    const float* x         = (const float*)d_in[0];
    const float* weights   = (const float*)d_in[1];
    const float* biases    = (const float*)d_in[2];
    const int*   edge_from = (const int*)d_in[3];
    // d_in[4] = edge_to: structured (repeat), not needed.
    float* out = (float*)d_out;
    float* ws  = (float*)d_ws;   // 6 hidden-level activations, 4096 floats each (96 KB)

    constexpr int NLVL = 8;
    constexpr int LS[NLVL] = {1024, 4096, 4096, 4096, 4096, 4096, 4096, 512};
    int starts[NLVL + 1];
    starts[0] = 0;
    for (int i = 0; i < NLVL; ++i) starts[i + 1] = starts[i] + LS[i];

    size_t eoff = 0;
    const float* src = x;
    for (int l = 0; l < NLVL - 1; ++l) {
        const int src_n = LS[l];
        const int tgt_n = LS[l + 1];
        float* dst;
        const float* bias = nullptr;
        if (l == NLVL - 2) { dst = out; bias = biases + starts[NLVL - 1]; }
        else               { dst = ws + (size_t)l * 4096; }

        dim3 grid(tgt_n / TGT_PER_BLOCK);
        hipLaunchKernelGGL(nn_level_kernel, grid, dim3(BLOCK_THREADS), 0, stream,
                           src, weights + eoff, edge_from + eoff, dst, bias,
                           starts[l], src_n);
        src = dst;
        eoff += (size_t)tgt_n * KK;
    }
}